// SpanRepresentation_12687333392637
// MI455X (gfx1250) — compile-verified
//
#include <hip/hip_runtime.h>
#include <stdint.h>

// Problem constants (fixed by the reference harness).
#define BATCH     8
#define SEQLEN    512
#define DIM       768
#define WEMB      64
#define MAXSPAN   10
#define NSPANS    5075            // sum_{w=1..10} (512 - w + 1)
#define ROW_F4    784             // (4*768 + 64) / 4 float4 chunks per row
#define SEG_F4    192             // 768 / 4
#define ROW_FLOATS 3136           // 4*768 + 64

// ---- CDNA5 async LDS<->global copy primitives (ASYNCcnt-tracked) ----------
__device__ __forceinline__ void async_load_b128(uint32_t lds_off, const void* gsrc) {
    // LDS byte offset comes from the VDST operand; per-lane 16B global->LDS, no VGPR data.
    // Default RT temporal hint: gather sources (features/pos, ~25MB) are
    // reused ~20x per position and should stay resident in the 192MB L2.
    asm volatile("global_load_async_to_lds_b128 %0, %1, off"
                 :: "v"(lds_off), "v"((uint64_t)(uintptr_t)gsrc)
                 : "memory");
}
__device__ __forceinline__ void async_store_b128(void* gdst, uint32_t lds_off) {
    // Non-temporal store: ~509MB of streamed output is never re-read on
    // device; keep it from churning L2 so the gather reads stay cached.
    asm volatile("global_store_async_from_lds_b128 %0, %1, off th:TH_STORE_NT"
                 :: "v"((uint64_t)(uintptr_t)gdst), "v"(lds_off)
                 : "memory");
}
__device__ __forceinline__ void wait_async0() {
    asm volatile("s_wait_asynccnt 0" ::: "memory");
}

__global__ __launch_bounds__(256) void span_gather_async_kernel(
    const float* __restrict__ feat,   // [B, L, D]
    const float* __restrict__ pos,    // [B, L, D]
    const float* __restrict__ wemb,   // [17, WE]
    float* __restrict__ out)          // [B, S, 4D + WE]
{
    __shared__ float4 smem[ROW_F4];   // 12544 B staging for one output row

    const int s = blockIdx.x;         // span index, 0..NSPANS-1
    const int b = blockIdx.y;         // batch index

    // span index -> (width w, start): widths enumerated 1..10, each with
    // (L - w + 1) consecutive starts.
    int rem = s, w = 1, cnt = SEQLEN;
    while (rem >= cnt) { rem -= cnt; ++w; --cnt; }
    const int start = rem;
    const int end   = start + w - 1;

    // bucket = searchsorted(bins, w, side='right') - 1
    const int bins[17] = {0,1,2,3,4,5,6,7,8,9,10,15,20,25,30,50,80};
    int bucket = 0;
#pragma unroll
    for (int i = 1; i < 17; ++i) bucket += (w >= bins[i]) ? 1 : 0;

    const float* fs = feat + ((size_t)b * SEQLEN + start) * DIM;
    const float* fe = feat + ((size_t)b * SEQLEN + end)   * DIM;
    const float* ps = pos  + ((size_t)b * SEQLEN + start) * DIM;
    const float* pe = pos  + ((size_t)b * SEQLEN + end)   * DIM;
    const float* wr = wemb + (size_t)bucket * WEMB;
    float* orow = out + ((size_t)b * NSPANS + s) * ROW_FLOATS;

    // Low 32 bits of a flat shared pointer == LDS byte offset (aperture base
    // has zero low bits), usable directly as the async op's VDST/VSRC offset.
    const uint32_t lds_base = (uint32_t)(size_t)&smem[0];

    // Stage the gathered row into LDS with async b128 copies (fully
    // coalesced: lane k of each wave handles chunk base+k -> 512 B/wave/op).
    for (int i = threadIdx.x; i < ROW_F4; i += 256) {
        const float* src;
        if      (i <     SEG_F4) src = fs + 4 * i;
        else if (i < 2 * SEG_F4) src = fe + 4 * (i -     SEG_F4);
        else if (i < 3 * SEG_F4) src = ps + 4 * (i - 2 * SEG_F4);
        else if (i < 4 * SEG_F4) src = pe + 4 * (i - 3 * SEG_F4);
        else                     src = wr + 4 * (i - 4 * SEG_F4);
        async_load_b128(lds_base + 16u * (uint32_t)i, src);
    }
    wait_async0();   // loads land in LDS (in-order done per wave)

    // Stream the row to HBM straight out of LDS (no VGPR round trip). Each
    // wave only touches slots it loaded itself, so no barrier is required.
    for (int i = threadIdx.x; i < ROW_F4; i += 256) {
        async_store_b128(orow + 4 * i, lds_base + 16u * (uint32_t)i);
    }
    wait_async0();   // drain before LDS deallocation (ENDPGM also waits)
}

extern "C" void kernel_launch(void* const* d_in, const int* in_sizes, int n_in,
                              void* d_out, int out_size, void* d_ws, size_t ws_size,
                              hipStream_t stream) {
    (void)in_sizes; (void)n_in; (void)out_size; (void)d_ws; (void)ws_size;
    const float* feat = (const float*)d_in[0];
    const float* pos  = (const float*)d_in[1];
    const float* wemb = (const float*)d_in[2];
    // d_in[3] = batch_max_seq_len (== SEQLEN, fixed by the harness)
    float* out = (float*)d_out;

    dim3 grid(NSPANS, BATCH);
    dim3 block(256);
    hipLaunchKernelGGL(span_gather_async_kernel, grid, block, 0, stream,
                       feat, pos, wemb, out);
}